// SphericalConvBlock_5600637354410
// MI455X (gfx1250) — compile-verified
//
#include <hip/hip_runtime.h>
#include <hip/hip_bf16.h>

typedef __attribute__((ext_vector_type(2))) float v2f;
typedef __attribute__((ext_vector_type(8))) float v8f;

#define V_N   196608
#define B_N   2
#define M_N   (V_N * B_N)     // 393216 rows for the dense GEMMs
#define DEG   9
#define EPSI  1e-5

// ---------------------------------------------------------------------------
// [B,V,16] -> [M=V*B, 16]  (per-vertex contiguous layout for the SpMV gathers)
// ---------------------------------------------------------------------------
__global__ void k_transpose_in(const float* __restrict__ in, float* __restrict__ out) {
    int idx = blockIdx.x * blockDim.x + threadIdx.x;   // over B*V*4 float4s
    int f4 = idx & 3;
    int v  = (idx >> 2) % V_N;
    int b  = idx / (V_N * 4);
    float4 val = ((const float4*)in)[((size_t)b * V_N + v) * 4 + f4];
    ((float4*)out)[((size_t)v * B_N + b) * 4 + f4] = val;
}

// ---------------------------------------------------------------------------
// Fixed-degree-9 SpMV over [V, BF] features.  y = alpha * (L x) (- z if HASZ)
// One wave per vertex, lane owns VEC contiguous channels (BF = 32*VEC).
// Each edge gather is a contiguous 128/256B chunk -> L2-resident.
// ---------------------------------------------------------------------------
template <int VEC, bool HASZ>
__global__ void k_spmv(float* __restrict__ y, const float* __restrict__ x,
                       const long long* __restrict__ cols,
                       const float* __restrict__ vals,
                       const float* __restrict__ z, float alpha) {
    const int BF = 32 * VEC;
    int v    = blockIdx.x * 8 + (threadIdx.x >> 5);
    int lane = threadIdx.x & 31;
    long long e = (long long)v * DEG;

    float acc[VEC];
#pragma unroll
    for (int t = 0; t < VEC; ++t) acc[t] = 0.f;

#pragma unroll
    for (int j = 0; j < DEG; ++j) {
        long long c = cols[e + j];                    // int64 indices per reference
        float     w = vals[e + j];
        const float* xp = x + (size_t)c * BF + lane * VEC;
        if constexpr (VEC == 2) {
            float2 xv = *(const float2*)xp;
            acc[0] = fmaf(w, xv.x, acc[0]);
            acc[1] = fmaf(w, xv.y, acc[1]);
        } else {
            acc[0] = fmaf(w, xp[0], acc[0]);
        }
    }

    float* yp = y + (size_t)v * BF + lane * VEC;
    if constexpr (HASZ) {
        const float* zp = z + (size_t)v * BF + lane * VEC;
#pragma unroll
        for (int t = 0; t < VEC; ++t) yp[t] = fmaf(alpha, acc[t], -zp[t]);
    } else {
#pragma unroll
        for (int t = 0; t < VEC; ++t) yp[t] = alpha * acc[t];
    }
}

// ---------------------------------------------------------------------------
// Chebyshev GEMM via V_WMMA_F32_16X16X4_F32 (fp32-exact matrix core path).
// out[r, 0:32] = sum_k x_k[r, :] @ W[k]  with A rows = concat(x0,x1,x2).
// One wave computes a 16x32 tile: 2 accumulators, KT = 3*FIN/4 k-steps.
// A fragment layout (16x4 f32): lanes 0-15 hold K=0,1; lanes 16-31 hold K=2,3.
// B fragment layout (4x16 f32): VGPR i holds row K=2*khalf+i, N=lane&15.
// TRANSOUT=1 writes [B,V,32] (final output layout) instead of [M,32].
// ---------------------------------------------------------------------------
template <int FIN, int TRANSOUT>
__global__ void k_cheb_gemm(const float* __restrict__ x0, const float* __restrict__ x1,
                            const float* __restrict__ x2, const float* __restrict__ W,
                            float* __restrict__ out) {
    const int KD = 3 * FIN;
    const int KT = KD / 4;
    __shared__ float sW[KD * 32];
    for (int i = threadIdx.x; i < KD * 32; i += 256) sW[i] = W[i];
    __syncthreads();

    int wave  = blockIdx.x * 8 + (threadIdx.x >> 5);
    int lane  = threadIdx.x & 31;
    int mlan  = lane & 15;
    int khalf = lane >> 4;
    int row0  = wave * 16;

    const float* bufs[3] = {x0, x1, x2};
    v8f c0 = {};
    v8f c1 = {};

#pragma unroll
    for (int kt = 0; kt < KT; ++kt) {
        const int buf = (kt * 4) / FIN;          // FIN is a multiple of 4:
        int f     = kt * 4 + khalf * 2;          // both halves stay in one buffer
        int local = f - buf * FIN;
        v2f a = *(const v2f*)(bufs[buf] + (size_t)(row0 + mlan) * FIN + local);
        v2f b0, b1;
        b0.x = sW[(f + 0) * 32 + mlan];
        b0.y = sW[(f + 1) * 32 + mlan];
        b1.x = sW[(f + 0) * 32 + 16 + mlan];
        b1.y = sW[(f + 1) * 32 + 16 + mlan];
        c0 = __builtin_amdgcn_wmma_f32_16x16x4_f32(false, a, false, b0, (short)0, c0, false, false);
        c1 = __builtin_amdgcn_wmma_f32_16x16x4_f32(false, a, false, b1, (short)0, c1, false, false);
    }

#pragma unroll
    for (int j = 0; j < 8; ++j) {
        int r = row0 + j + khalf * 8;            // D layout: VGPR j -> M=j / j+8
        size_t o0, o1;
        if constexpr (TRANSOUT) {
            int vv = r >> 1, bb = r & 1;         // r = v*B + b, B = 2
            size_t base = ((size_t)bb * V_N + vv) * 32;
            o0 = base + mlan;
            o1 = base + 16 + mlan;
        } else {
            o0 = (size_t)r * 32 + mlan;
            o1 = (size_t)r * 32 + 16 + mlan;
        }
        out[o0] = c0[j];
        out[o1] = c1[j];
    }
}

// ---------------------------------------------------------------------------
// BatchNorm stats, deterministic: each block writes its own partial slot
// (no atomics, no init needed), reduced later in fixed order.
// ---------------------------------------------------------------------------
__global__ void k_bn_stats(const float* __restrict__ h, double* __restrict__ part) {
    int c = threadIdx.x & 31;
    int g = threadIdx.x >> 5;
    double s = 0.0, s2 = 0.0;
    for (int r = blockIdx.x * 8 + g; r < M_N; r += gridDim.x * 8) {
        double xd = (double)h[(size_t)r * 32 + c];
        s += xd;
        s2 += xd * xd;
    }
    __shared__ double sh[8][64];
    sh[g][c] = s;
    sh[g][32 + c] = s2;
    __syncthreads();
    if (g == 0) {
        for (int t = 1; t < 8; ++t) { s += sh[t][c]; s2 += sh[t][32 + c]; }
        part[(size_t)blockIdx.x * 64 + c]      = s;
        part[(size_t)blockIdx.x * 64 + 32 + c] = s2;
    }
}

// Reduce 256 partials per channel in fixed order -> per-channel scale/shift.
__global__ void k_bn_prep(const double* __restrict__ part, const float* __restrict__ gamma,
                          const float* __restrict__ beta, float* __restrict__ ss) {
    int c = threadIdx.x;  // 32 threads = one wave
    double s = 0.0, s2 = 0.0;
    for (int b = 0; b < 256; ++b) {
        s  += part[(size_t)b * 64 + c];
        s2 += part[(size_t)b * 64 + 32 + c];
    }
    double mean = s / (double)M_N;
    double var  = s2 / (double)M_N - mean * mean;     // biased var, as reference
    double inv  = 1.0 / sqrt(var + (double)EPSI);
    double sc   = (double)gamma[c] * inv;
    ss[c]      = (float)sc;
    ss[32 + c] = (float)((double)beta[c] - mean * sc);
}

// y = relu(x * scale[c] + shift[c]), vectorized float4 (channels are last dim=32).
__global__ void k_bn_apply(float* __restrict__ h, const float* __restrict__ ss) {
    int idx = blockIdx.x * blockDim.x + threadIdx.x;  // over M*8 float4s
    int c4 = idx & 7;
    float4 sc = ((const float4*)ss)[c4];
    float4 sf = ((const float4*)ss)[8 + c4];
    float4 x  = ((float4*)h)[idx];
    x.x = fmaxf(fmaf(x.x, sc.x, sf.x), 0.f);
    x.y = fmaxf(fmaf(x.y, sc.y, sf.y), 0.f);
    x.z = fmaxf(fmaf(x.z, sc.z, sf.z), 0.f);
    x.w = fmaxf(fmaf(x.w, sc.w, sf.w), 0.f);
    ((float4*)h)[idx] = x;
}

// ---------------------------------------------------------------------------
extern "C" void kernel_launch(void* const* d_in, const int* in_sizes, int n_in,
                              void* d_out, int out_size, void* d_ws, size_t ws_size,
                              hipStream_t stream) {
    const float*     x    = (const float*)d_in[0];
    /* d_in[1] = rows: unused, graph has fixed degree 9 with sorted rows   */
    const long long* cols = (const long long*)d_in[2];
    const float*     vals = (const float*)d_in[3];
    const float*     W1   = (const float*)d_in[4];
    const float*     g1   = (const float*)d_in[5];
    const float*     b1   = (const float*)d_in[6];
    const float*     W2   = (const float*)d_in[7];
    const float*     g2   = (const float*)d_in[8];
    const float*     b2   = (const float*)d_in[9];
    float* out = (float*)d_out;

    char* ws = (char*)d_ws;
    const size_t SZ32 = (size_t)M_N * 32 * sizeof(float);   // 50.3 MB
    const size_t SZ16 = (size_t)M_N * 16 * sizeof(float);   // 25.2 MB
    float*  R0   = (float*)(ws);                            // L1 x0 / L2 x1
    float*  R1   = (float*)(ws + SZ32);                     // L1 x1 / L2 x2
    float*  R2   = (float*)(ws + 2 * SZ32);                 // L1 x2
    float*  R3   = (float*)(ws + 2 * SZ32 + SZ16);          // h1 = L2 x0
    double* part = (double*)(ws + 3 * SZ32 + SZ16);         // 256*64 doubles
    float*  ss   = (float*)(ws + 3 * SZ32 + SZ16 + 256 * 64 * sizeof(double));

    // ---- Layer 1 (Fin = 16) ----
    k_transpose_in<<<(B_N * V_N * 4) / 256, 256, 0, stream>>>(x, R0);
    k_spmv<1, false><<<V_N / 8, 256, 0, stream>>>(R1, R0, cols, vals, nullptr, 1.0f);
    k_spmv<1, true ><<<V_N / 8, 256, 0, stream>>>(R2, R1, cols, vals, R0, 2.0f);
    k_cheb_gemm<16, 0><<<M_N / 128, 256, 0, stream>>>(R0, R1, R2, W1, R3);
    k_bn_stats<<<256, 256, 0, stream>>>(R3, part);
    k_bn_prep<<<1, 32, 0, stream>>>(part, g1, b1, ss);
    k_bn_apply<<<(M_N * 8) / 256, 256, 0, stream>>>(R3, ss);

    // ---- Layer 2 (Fin = 32), GEMM writes final [B,V,32] layout into d_out ----
    k_spmv<2, false><<<V_N / 8, 256, 0, stream>>>(R0, R3, cols, vals, nullptr, 1.0f);
    k_spmv<2, true ><<<V_N / 8, 256, 0, stream>>>(R1, R0, cols, vals, R3, 2.0f);
    k_cheb_gemm<32, 1><<<M_N / 128, 256, 0, stream>>>(R3, R0, R1, W2, out);
    k_bn_stats<<<256, 256, 0, stream>>>(out, part);
    k_bn_prep<<<1, 32, 0, stream>>>(part, g2, b2, ss);
    k_bn_apply<<<(M_N * 8) / 256, 256, 0, stream>>>(out, ss);
}